// PainnModel_1511828488746
// MI455X (gfx1250) — compile-verified
//
#include <hip/hip_runtime.h>
#include <hip/hip_bf16.h>
#include <math.h>

// ---------------------------------------------------------------------------
// PaiNN energy + forces on MI455X (gfx1250, wave32, WMMA).
//
//  - All linears via V_WMMA_F32_16X16X4_F32 (branchless 16x32-strip GEMM).
//  - Filter path fused: the E x 384 "fwlin" tensor is never materialized.
//    Edge kernels compute it in-register with WMMA (A = 16x20 rbf rows per
//    wave, hoisted; 16-column strips -> only 3 live accumulators, no spills)
//    and feed accumulator lanes straight into the gather/scatter math.
//  - Edge streams are the 23.3 TB/s roofline term -> each remaining stream
//    is touched exactly once; segment sums via global_atomic_add_f32.
//  - Backward needs no weight grads => every linear backward is dY @ W^T
//    (TRANSW template path). Workspace ~900 MB.
// ---------------------------------------------------------------------------

#define NB   8
#define NA   8192        // total atoms (B*A)
#define NE   393216      // edges
#define HD   128
#define H3   384
#define NL   3
#define RBFN 20
#define FCUT 5.0f
#define PIF  3.14159265358979323846f

typedef float v2f __attribute__((ext_vector_type(2)));
typedef float v8f __attribute__((ext_vector_type(8)));

__device__ __forceinline__ float sigm(float x) { return 1.f / (1.f + __expf(-x)); }
__device__ __forceinline__ float silu_f(float x) { return x * sigm(x); }
__device__ __forceinline__ float silu_g(float x) { float s = sigm(x); return s * (1.f + x * (1.f - s)); }

__device__ __forceinline__ v8f wmma4(v2f a, v2f b, v8f c)
{
    return __builtin_amdgcn_wmma_f32_16x16x4_f32(false, a, false, b, (short)0, c,
                                                 false, false);
}

// ---------------------------------------------------------------------------
// WMMA f32 GEMM:  C[M,Nc] = A[M,K] @ B + bias   (B = W[K,Nc] or W[Nc,K]^T)
// flags: bit1 = accumulate into C (bias ignored then).
// Requirements (hold for every call): M % 16 == 0, K % 4 == 0, ntN even.
// One wave -> 16x32 C strip (two WMMA tiles sharing the A fragment).
// ---------------------------------------------------------------------------
template <int K, bool TRANSW>
__global__ void k_gemm_t(const float* __restrict__ A, const float* __restrict__ W,
                         const float* __restrict__ bias, float* __restrict__ C,
                         int M, int Nc, int flags)
{
    const int lane = threadIdx.x & 31;
    const int wave = threadIdx.x >> 5;
    const long pair = (long)blockIdx.x * (blockDim.x >> 5) + wave;
    const int ntNp = ((Nc + 15) >> 4) >> 1;   // pairs of N tiles (ntN even)
    const int ntM  = M >> 4;
    if (pair >= (long)ntNp * ntM) return;
    const int tM  = (int)(pair / ntNp);
    const int tNp = (int)(pair % ntNp);
    const int row0 = tM << 4;
    const int col0 = tNp << 5;
    const int half = lane >> 4, l16 = lane & 15;
    const int c0 = col0 + l16, c1 = c0 + 16;
    const bool ok0 = (c0 < Nc), ok1 = (c1 < Nc);
    const bool accum = (flags & 2) != 0;

    v8f acc0, acc1;
    if (accum) {
        #pragma unroll
        for (int i = 0; i < 8; ++i) {
            int r = row0 + i + half * 8;
            acc0[i] = ok0 ? C[(long)r * Nc + c0] : 0.f;
            acc1[i] = ok1 ? C[(long)r * Nc + c1] : 0.f;
        }
    } else if (bias) {
        float bv0 = ok0 ? bias[c0] : 0.f;
        float bv1 = ok1 ? bias[c1] : 0.f;
        #pragma unroll
        for (int i = 0; i < 8; ++i) { acc0[i] = bv0; acc1[i] = bv1; }
    } else {
        #pragma unroll
        for (int i = 0; i < 8; ++i) { acc0[i] = 0.f; acc1[i] = 0.f; }
    }

    const int rA = row0 + l16;
    const float* pA = A + (long)rA * K + half * 2;

    if (col0 + 32 <= Nc) {                 // full strip: no guards at all
        if (!TRANSW) {
            const float* pB = W + (long)(half * 2) * Nc + c0;
            #pragma unroll 8
            for (int kk = 0; kk < K; kk += 4) {
                v2f a = *(const v2f*)pA;
                v2f b0, b1;
                b0.x = pB[0];  b0.y = pB[Nc];
                b1.x = pB[16]; b1.y = pB[Nc + 16];
                acc0 = wmma4(a, b0, acc0);
                acc1 = wmma4(a, b1, acc1);
                pA += 4; pB += 4 * (long)Nc;
            }
        } else {
            const float* pB0 = W + (long)c0 * K + half * 2;
            const float* pB1 = W + (long)c1 * K + half * 2;
            #pragma unroll 8
            for (int kk = 0; kk < K; kk += 4) {
                v2f a  = *(const v2f*)pA;
                v2f b0 = *(const v2f*)pB0;
                v2f b1 = *(const v2f*)pB1;
                acc0 = wmma4(a, b0, acc0);
                acc1 = wmma4(a, b1, acc1);
                pA += 4; pB0 += 4; pB1 += 4;
            }
        }
    } else {                               // partial columns (Nc==20 GEMMs)
        const int c0c = ok0 ? c0 : (Nc - 1);
        const int c1c = ok1 ? c1 : (Nc - 1);
        if (!TRANSW) {
            const float* pB = W + (long)(half * 2) * Nc;
            #pragma unroll 4
            for (int kk = 0; kk < K; kk += 4) {
                v2f a = *(const v2f*)pA;
                v2f b0, b1;
                b0.x = ok0 ? pB[c0c]      : 0.f;
                b0.y = ok0 ? pB[Nc + c0c] : 0.f;
                b1.x = ok1 ? pB[c1c]      : 0.f;
                b1.y = ok1 ? pB[Nc + c1c] : 0.f;
                acc0 = wmma4(a, b0, acc0);
                acc1 = wmma4(a, b1, acc1);
                pA += 4; pB += 4 * (long)Nc;
            }
        } else {
            const float* pB0 = W + (long)c0c * K + half * 2;
            const float* pB1 = W + (long)c1c * K + half * 2;
            #pragma unroll 4
            for (int kk = 0; kk < K; kk += 4) {
                v2f a  = *(const v2f*)pA;
                v2f b0 = *(const v2f*)pB0;
                v2f b1 = *(const v2f*)pB1;
                if (!ok0) { b0.x = 0.f; b0.y = 0.f; }
                if (!ok1) { b1.x = 0.f; b1.y = 0.f; }
                acc0 = wmma4(a, b0, acc0);
                acc1 = wmma4(a, b1, acc1);
                pA += 4; pB0 += 4; pB1 += 4;
            }
        }
    }

    #pragma unroll
    for (int i = 0; i < 8; ++i) {
        int r = row0 + i + half * 8;
        if (ok0) C[(long)r * Nc + c0] = acc0[i];
        if (ok1) C[(long)r * Nc + c1] = acc1[i];
    }
}

// ---------------------------------------------------------------------------
// Fused edge forward: one wave owns 16 edges. fwlin = rbf@Fw + Fb is computed
// in-register via WMMA (K=20, A-fragments hoisted) per 16-column strip; the
// 3 chunk accumulators feed the message construction + scatter directly.
// ---------------------------------------------------------------------------
__global__ void k_edge_fwd_fused(const float* __restrict__ rbf, const float* __restrict__ Fw,
                                 const float* __restrict__ Fb, const float* __restrict__ fcut,
                                 const float* __restrict__ h, const float* __restrict__ v_in,
                                 const float* __restrict__ unit, const int* __restrict__ edge,
                                 float* __restrict__ s, float* __restrict__ v)
{
    const int lane = threadIdx.x & 31;
    const int wave = threadIdx.x >> 5;
    const int e0 = (blockIdx.x * (blockDim.x >> 5) + wave) << 4;   // 16 edges/wave
    if (e0 >= NE) return;                       // wave-uniform
    const int half = lane >> 4, l16 = lane & 15;

    v2f afrag[5];                               // 16x20 rbf rows, loaded once
    {
        const float* pA0 = rbf + (long)(e0 + l16) * RBFN + half * 2;
        #pragma unroll
        for (int k5 = 0; k5 < 5; ++k5) afrag[k5] = *(const v2f*)(pA0 + k5 * 4);
    }

    for (int js = 0; js < 128; js += 16) {
        const int jL = js + l16;
        v8f acc[3];                             // one per 128-chunk
        #pragma unroll
        for (int q = 0; q < 3; ++q) {
            float bv = Fb[q * 128 + jL];
            #pragma unroll
            for (int i = 0; i < 8; ++i) acc[q][i] = bv;
        }
        #pragma unroll
        for (int k5 = 0; k5 < 5; ++k5) {
            const float* pB = Fw + (long)(k5 * 4 + half * 2) * H3 + jL;
            #pragma unroll
            for (int q = 0; q < 3; ++q) {
                v2f b; b.x = pB[q * 128]; b.y = pB[H3 + q * 128];
                acc[q] = wmma4(afrag[k5], b, acc[q]);
            }
        }
        // consume: lane holds (edge = e0+i+half*8, column j = jL) per VGPR i
        #pragma unroll
        for (int i = 0; i < 8; ++i) {
            const int e = e0 + i + half * 8;
            const float fc = fcut[e];
            const int dst = edge[2 * e], src = edge[2 * e + 1];
            const long sb = (long)src * H3;
            const int j = jL;
            float gsv = acc[0][i] * fc * h[sb + j];
            float gev = acc[1][i] * fc * h[sb + 128 + j];
            float msc = acc[2][i] * fc * h[sb + 256 + j];
            atomicAdd(&s[(long)dst * HD + j], msc);
            #pragma unroll
            for (int c = 0; c < 3; ++c) {
                float mv = v_in[((long)src * 3 + c) * HD + j] * gsv
                         + gev * unit[3 * e + c];
                atomicAdd(&v[((long)dst * 3 + c) * HD + j], mv);
            }
        }
    }
}

// ---------------------------------------------------------------------------
// Fused edge backward: recomputes fwlin in-register via WMMA (rbf is 31MB and
// L2-resident), then runs the full message backward per lane. Emits d_fwlin
// (needed by the d_rbf GEMM), d_h, d_v_in, d_fcut, d_unit.
// ---------------------------------------------------------------------------
__global__ void k_edge_bwd_fused(const float* __restrict__ rbf, const float* __restrict__ Fw,
                                 const float* __restrict__ Fb, const float* __restrict__ fcut,
                                 const float* __restrict__ h, const float* __restrict__ v_in,
                                 const float* __restrict__ unit, const int* __restrict__ edge,
                                 const float* __restrict__ dS, const float* __restrict__ dV,
                                 float* __restrict__ dV2, float* __restrict__ dH,
                                 float* __restrict__ dFwlin, float* __restrict__ dFc,
                                 float* __restrict__ dUn)
{
    const int lane = threadIdx.x & 31;
    const int wave = threadIdx.x >> 5;
    const int e0 = (blockIdx.x * (blockDim.x >> 5) + wave) << 4;
    if (e0 >= NE) return;
    const int half = lane >> 4, l16 = lane & 15;

    v2f afrag[5];
    {
        const float* pA0 = rbf + (long)(e0 + l16) * RBFN + half * 2;
        #pragma unroll
        for (int k5 = 0; k5 < 5; ++k5) afrag[k5] = *(const v2f*)(pA0 + k5 * 4);
    }

    for (int js = 0; js < 128; js += 16) {
        const int jL = js + l16;
        v8f acc[3];
        #pragma unroll
        for (int q = 0; q < 3; ++q) {
            float bv = Fb[q * 128 + jL];
            #pragma unroll
            for (int i = 0; i < 8; ++i) acc[q][i] = bv;
        }
        #pragma unroll
        for (int k5 = 0; k5 < 5; ++k5) {
            const float* pB = Fw + (long)(k5 * 4 + half * 2) * H3 + jL;
            #pragma unroll
            for (int q = 0; q < 3; ++q) {
                v2f b; b.x = pB[q * 128]; b.y = pB[H3 + q * 128];
                acc[q] = wmma4(afrag[k5], b, acc[q]);
            }
        }
        #pragma unroll
        for (int i = 0; i < 8; ++i) {
            const int e = e0 + i + half * 8;
            const float fc = fcut[e];
            const int dst = edge[2 * e], src = edge[2 * e + 1];
            const long eb = (long)e * H3, sb = (long)src * H3;
            const int j = jL;
            float fl0 = acc[0][i], fl1 = acc[1][i], fl2 = acc[2][i];
            float fw0 = fl0 * fc, fw1 = fl1 * fc, fw2 = fl2 * fc;
            float h0 = h[sb + j], h1 = h[sb + 128 + j], h2 = h[sb + 256 + j];
            float gsv = fw0 * h0, gev = fw1 * h1;
            float dmsc = dS[(long)dst * HD + j];
            float dgsv = 0.f, dgev = 0.f;
            #pragma unroll
            for (int c = 0; c < 3; ++c) {
                float dmv = dV[((long)dst * 3 + c) * HD + j];
                float vi  = v_in[((long)src * 3 + c) * HD + j];
                dgsv += dmv * vi;
                dgev += dmv * unit[3 * e + c];
                atomicAdd(&dV2[((long)src * 3 + c) * HD + j], dmv * gsv);
                atomicAdd(&dUn[3 * e + c], dmv * gev);
            }
            atomicAdd(&dH[sb + j],       dgsv * fw0);
            atomicAdd(&dH[sb + 128 + j], dgev * fw1);
            atomicAdd(&dH[sb + 256 + j], dmsc * fw2);
            float dfw0 = dgsv * h0, dfw1 = dgev * h1, dfw2 = dmsc * h2;
            dFwlin[eb + j]       = dfw0 * fc;
            dFwlin[eb + 128 + j] = dfw1 * fc;
            dFwlin[eb + 256 + j] = dfw2 * fc;
            atomicAdd(&dFc[e], dfw0 * fl0 + dfw1 * fl1 + dfw2 * fl2);
        }
    }
}

// --------------------------- elementwise kernels ---------------------------

__global__ void k_silu(float* __restrict__ y, const float* __restrict__ x, long n)
{
    long t = (long)blockIdx.x * blockDim.x + threadIdx.x;
    if (t < n) y[t] = silu_f(x[t]);
}

__global__ void k_silu_bwd(float* __restrict__ g, const float* __restrict__ z, long n)
{
    long t = (long)blockIdx.x * blockDim.x + threadIdx.x;
    if (t < n) g[t] *= silu_g(z[t]);
}

__global__ void k_embed(float* __restrict__ s, const float* __restrict__ embed,
                        const int* __restrict__ elems)
{
    long t = (long)blockIdx.x * blockDim.x + threadIdx.x;
    if (t >= (long)NA * HD) return;
    int n = (int)(t >> 7), j = (int)(t & 127);
    s[t] = embed[(long)elems[n] * HD + j];
}

__global__ void k_edge_geom(const float* __restrict__ ed, float* __restrict__ dlen,
                            float* __restrict__ rbf, float* __restrict__ fcut,
                            float* __restrict__ unit)
{
    int e = blockIdx.x * blockDim.x + threadIdx.x;
    if (e >= NE) return;
    float x = ed[3 * e], y = ed[3 * e + 1], z = ed[3 * e + 2];
    float d = sqrtf(x * x + y * y + z * z);
    float inv = 1.f / d;
    dlen[e] = d;
    unit[3 * e] = x * inv; unit[3 * e + 1] = y * inv; unit[3 * e + 2] = z * inv;
    fcut[e] = (d < FCUT) ? 0.5f * (__cosf(PIF * d / FCUT) + 1.f) : 0.f;
    #pragma unroll
    for (int i = 0; i < RBFN; ++i)
        rbf[(long)e * RBFN + i] = __sinf(d * (float)(i + 1) * PIF / FCUT) * inv;
}

__global__ void k_vn_cat(const float* __restrict__ Vv, const float* __restrict__ s,
                         float* __restrict__ cat)
{
    long t = (long)blockIdx.x * blockDim.x + threadIdx.x;
    if (t >= (long)NA * HD) return;
    int n = (int)(t >> 7), j = (int)(t & 127);
    float s2 = 0.f;
    #pragma unroll
    for (int c = 0; c < 3; ++c) {
        float vv = Vv[((long)n * 3 + c) * HD + j];
        s2 += vv * vv;
    }
    cat[(long)n * 256 + j] = sqrtf(s2);
    cat[(long)n * 256 + 128 + j] = s[t];
}

__global__ void k_update(const float* __restrict__ m, const float* __restrict__ Uv,
                         const float* __restrict__ Vv, float* __restrict__ s,
                         float* __restrict__ v)
{
    long t = (long)blockIdx.x * blockDim.x + threadIdx.x;
    if (t >= (long)NA * HD) return;
    int n = (int)(t >> 7), j = (int)(t & 127);
    long mb = (long)n * H3;
    float a_vv = m[mb + j], a_sv = m[mb + 128 + j], a_ss = m[mb + 256 + j];
    float sdot = 0.f;
    #pragma unroll
    for (int c = 0; c < 3; ++c) {
        long vi = ((long)n * 3 + c) * HD + j;
        sdot += Uv[vi] * Vv[vi];
        v[vi] += a_vv * Uv[vi];
    }
    s[t] += a_sv * sdot + a_ss;
}

__global__ void k_energy(const float* __restrict__ h1, const float* __restrict__ r2w,
                         const float* __restrict__ r2b, const int* __restrict__ image_idx,
                         float* __restrict__ out)
{
    int n = blockIdx.x * blockDim.x + threadIdx.x;
    if (n >= NA) return;
    float acc = r2b[0];
    for (int j = 0; j < HD; ++j) acc += h1[(long)n * HD + j] * r2w[j];
    atomicAdd(&out[image_idx[n]], acc);
}

// d(e_atom)/d(z1) = r2w * silu'(z1)   (dE/de_atom = STD = 1)
__global__ void k_dz1(const float* __restrict__ z1, const float* __restrict__ r2w,
                      float* __restrict__ g)
{
    long t = (long)blockIdx.x * blockDim.x + threadIdx.x;
    if (t >= (long)NA * HD) return;
    int j = (int)(t & 127);
    g[t] = r2w[j] * silu_g(z1[t]);
}

// update-block backward, part 1: d_m, d_Uv, d_Vv (a_vv / sdot paths)
__global__ void k_upd_bwd1(const float* __restrict__ m, const float* __restrict__ Uv,
                           const float* __restrict__ Vv, const float* __restrict__ dS,
                           const float* __restrict__ dV, float* __restrict__ dM,
                           float* __restrict__ dUv, float* __restrict__ dVv)
{
    long t = (long)blockIdx.x * blockDim.x + threadIdx.x;
    if (t >= (long)NA * HD) return;
    int n = (int)(t >> 7), j = (int)(t & 127);
    long mb = (long)n * H3;
    float a_vv = m[mb + j], a_sv = m[mb + 128 + j];
    float ds = dS[t];
    float dsd = ds * a_sv;
    float sdot = 0.f, d_avv = 0.f;
    #pragma unroll
    for (int c = 0; c < 3; ++c) {
        long vi = ((long)n * 3 + c) * HD + j;
        float u = Uv[vi], vv = Vv[vi], dv = dV[vi];
        sdot += u * vv;
        d_avv += dv * u;
        dUv[vi] = dv * a_vv + dsd * vv;
        dVv[vi] = dsd * u;
    }
    dM[mb + j] = d_avv;
    dM[mb + 128 + j] = ds * sdot;
    dM[mb + 256 + j] = ds;
}

// update-block backward, part 2: Vn grad into dVv, cat grad into dS
__global__ void k_upd_bwd2(const float* __restrict__ Vv, const float* __restrict__ dCat,
                           float* __restrict__ dVv, float* __restrict__ dS)
{
    long t = (long)blockIdx.x * blockDim.x + threadIdx.x;
    if (t >= (long)NA * HD) return;
    int n = (int)(t >> 7), j = (int)(t & 127);
    float vv[3]; float s2 = 0.f;
    #pragma unroll
    for (int c = 0; c < 3; ++c) {
        vv[c] = Vv[((long)n * 3 + c) * HD + j];
        s2 += vv[c] * vv[c];
    }
    float vn = sqrtf(s2);
    float dvn = dCat[(long)n * 256 + j];
    float r = (vn > 0.f) ? dvn / vn : 0.f;
    #pragma unroll
    for (int c = 0; c < 3; ++c) dVv[((long)n * 3 + c) * HD + j] += r * vv[c];
    dS[t] += dCat[(long)n * 256 + 128 + j];
}

// geometry backward + force scatter
__global__ void k_geom_bwd(const float* __restrict__ dlen, const float* __restrict__ unit,
                           const float* __restrict__ dRbf, const float* __restrict__ dFc,
                           const float* __restrict__ dUn, const int* __restrict__ edge,
                           float* __restrict__ out)
{
    int e = blockIdx.x * blockDim.x + threadIdx.x;
    if (e >= NE) return;
    float d = dlen[e];
    float inv = 1.f / d;
    float g_d = 0.f;
    #pragma unroll
    for (int i = 0; i < RBFN; ++i) {
        float w = (float)(i + 1) * (PIF / FCUT);
        float arg = d * w;
        g_d += dRbf[(long)e * RBFN + i] * (w * __cosf(arg) * inv - __sinf(arg) * inv * inv);
    }
    if (d < FCUT) g_d += dFc[e] * (-0.5f * (PIF / FCUT) * __sinf(PIF * d / FCUT));
    float u0 = unit[3 * e], u1 = unit[3 * e + 1], u2 = unit[3 * e + 2];
    float g0 = dUn[3 * e], g1 = dUn[3 * e + 1], g2 = dUn[3 * e + 2];
    float dot = g0 * u0 + g1 * u1 + g2 * u2;
    float f0 = (g0 - u0 * dot) * inv + g_d * u0;
    float f1 = (g1 - u1 * dot) * inv + g_d * u1;
    float f2 = (g2 - u2 * dot) * inv + g_d * u2;
    int dst = edge[2 * e], src = edge[2 * e + 1];
    atomicAdd(&out[8 + (long)dst * 3 + 0],  f0);
    atomicAdd(&out[8 + (long)dst * 3 + 1],  f1);
    atomicAdd(&out[8 + (long)dst * 3 + 2],  f2);
    atomicAdd(&out[8 + (long)src * 3 + 0], -f0);
    atomicAdd(&out[8 + (long)src * 3 + 1], -f1);
    atomicAdd(&out[8 + (long)src * 3 + 2], -f2);
}

// ------------------------------- host side ---------------------------------

struct LinW { const float* w; const float* b; };

static inline float* carve(char*& p, size_t nfloats)
{
    float* r = (float*)p;
    p += ((nfloats * sizeof(float) + 255) / 256) * 256;
    return r;
}

static void launch_gemm(float* C, const float* A, const float* W, const float* bias,
                        int M, int Nc, int K, int flags, hipStream_t stream)
{
    long pairs = (long)((((Nc + 15) >> 4) >> 1)) * (M >> 4);
    int blocks = (int)((pairs + 7) / 8);
    bool tw = (flags & 1) != 0;
#define GEMM_CASE(KK)                                                             \
    case KK:                                                                      \
        if (tw) k_gemm_t<KK, true><<<blocks, 256, 0, stream>>>(A, W, bias, C, M, Nc, flags); \
        else    k_gemm_t<KK, false><<<blocks, 256, 0, stream>>>(A, W, bias, C, M, Nc, flags); \
        break;
    switch (K) {
        GEMM_CASE(128)
        GEMM_CASE(256)
        GEMM_CASE(384)
        default: break;
    }
#undef GEMM_CASE
}

extern "C" void kernel_launch(void* const* d_in, const int* in_sizes, int n_in,
                              void* d_out, int out_size, void* d_ws, size_t ws_size,
                              hipStream_t stream)
{
    (void)n_in; (void)ws_size;
    // --- map inputs (handles both insertion-order and sorted-key flattening) ---
    const int *elems, *edge, *image_idx;
    const float* edge_diff;
    bool topSorted = (in_sizes[0] == 2 * NE);   // sorted: edge first; insertion: elems first
    if (topSorted) {
        edge      = (const int*)d_in[0];
        edge_diff = (const float*)d_in[1];
        elems     = (const int*)d_in[2];
        image_idx = (const int*)d_in[3];
    } else {
        elems     = (const int*)d_in[0];
        edge      = (const int*)d_in[1];
        edge_diff = (const float*)d_in[2];
        image_idx = (const int*)d_in[3];
    }
    const float* embed = (const float*)d_in[5];

    // per-layer slot order: 0 filter, 1 smlp1, 2 smlp2, 3 U, 4 V, 5 umlp1, 6 umlp2
    LinW lay[NL][7]; LinW r1, r2;
    bool linSorted = (in_sizes[6] == HD);       // sorted: U.b(128) first; insertion: filter.w(7680)
    static const int permSort[7] = {3, 4, 0, 1, 2, 5, 6};
    int idx = 6;
    for (int l = 0; l < NL; ++l) {
        for (int i = 0; i < 7; ++i) {
            int slot = linSorted ? permSort[i] : i;
            if (linSorted) {
                lay[l][slot].b = (const float*)d_in[idx++];
                lay[l][slot].w = (const float*)d_in[idx++];
            } else {
                lay[l][slot].w = (const float*)d_in[idx++];
                lay[l][slot].b = (const float*)d_in[idx++];
            }
        }
    }
    if (linSorted) {
        r1.b = (const float*)d_in[idx++]; r1.w = (const float*)d_in[idx++];
        r2.b = (const float*)d_in[idx++]; r2.w = (const float*)d_in[idx++];
    } else {
        r1.w = (const float*)d_in[idx++]; r1.b = (const float*)d_in[idx++];
        r2.w = (const float*)d_in[idx++]; r2.b = (const float*)d_in[idx++];
    }

    // --- carve workspace ---
    char* p = (char*)d_ws;
    float* rbf  = carve(p, (size_t)NE * RBFN);
    float* dlen = carve(p, NE);
    float* fcut = carve(p, NE);
    float* unit = carve(p, (size_t)NE * 3);
    float* s    = carve(p, (size_t)NA * HD);
    float* v    = carve(p, (size_t)NA * H3);
    float *v_in[NL], *z_s1[NL], *h_st[NL], *Uv[NL], *Vv[NL], *z_u1[NL], *m_st[NL];
    for (int l = 0; l < NL; ++l) {
        v_in[l] = carve(p, (size_t)NA * H3);
        z_s1[l] = carve(p, (size_t)NA * HD);
        h_st[l] = carve(p, (size_t)NA * H3);
        Uv[l]   = carve(p, (size_t)NA * H3);
        Vv[l]   = carve(p, (size_t)NA * H3);
        z_u1[l] = carve(p, (size_t)NA * HD);
        m_st[l] = carve(p, (size_t)NA * H3);
    }
    float* z1    = carve(p, (size_t)NA * HD);
    float* tmpA  = carve(p, (size_t)NA * H3);
    float* catb  = carve(p, (size_t)NA * 256);
    float* d_s   = carve(p, (size_t)NA * HD);
    float* d_va  = carve(p, (size_t)NA * H3);
    float* d_vb  = carve(p, (size_t)NA * H3);
    float* d_m   = carve(p, (size_t)NA * H3);
    float* d_Uvb = carve(p, (size_t)NA * H3);
    float* d_Vvb = carve(p, (size_t)NA * H3);
    float* d_h   = carve(p, (size_t)NA * H3);
    float* d_rbf = carve(p, (size_t)NE * RBFN);
    float* d_fc  = carve(p, NE);
    float* d_un  = carve(p, (size_t)NE * 3);
    float* bufE2 = carve(p, (size_t)NE * H3);   // d_fwlin (feeds d_rbf GEMM)

    auto gemm = [&](float* C, const float* A, const float* W, const float* bias,
                    int M, int Nc, int K, int flags) {
        launch_gemm(C, A, W, bias, M, Nc, K, flags, stream);
    };
    auto gsz = [](long n) { return (unsigned)((n + 255) / 256); };
    const long NAH = (long)NA * HD;
    const int edgeBlocks = NE / 128;            // 8 waves/block, 16 edges/wave

    // ---------------------------- forward -----------------------------------
    hipMemsetAsync(d_out, 0, (size_t)out_size * sizeof(float), stream);
    hipMemsetAsync(v, 0, (size_t)NA * H3 * sizeof(float), stream);
    k_embed<<<gsz(NAH), 256, 0, stream>>>(s, embed, elems);
    k_edge_geom<<<gsz(NE), 256, 0, stream>>>(edge_diff, dlen, rbf, fcut, unit);

    for (int l = 0; l < NL; ++l) {
        gemm(z_s1[l], s, lay[l][1].w, lay[l][1].b, NA, HD, HD, 0);            // smlp1
        k_silu<<<gsz(NAH), 256, 0, stream>>>(tmpA, z_s1[l], NAH);
        gemm(h_st[l], tmpA, lay[l][2].w, lay[l][2].b, NA, H3, HD, 0);         // smlp2
        hipMemcpyAsync(v_in[l], v, (size_t)NA * H3 * sizeof(float),
                       hipMemcpyDeviceToDevice, stream);
        k_edge_fwd_fused<<<edgeBlocks, 256, 0, stream>>>(rbf, lay[l][0].w, lay[l][0].b,
                                                         fcut, h_st[l], v_in[l],
                                                         unit, edge, s, v);
        gemm(Uv[l], v, lay[l][3].w, lay[l][3].b, NA * 3, HD, HD, 0);          // U
        gemm(Vv[l], v, lay[l][4].w, lay[l][4].b, NA * 3, HD, HD, 0);          // V
        k_vn_cat<<<gsz(NAH), 256, 0, stream>>>(Vv[l], s, catb);
        gemm(z_u1[l], catb, lay[l][5].w, lay[l][5].b, NA, HD, 256, 0);        // umlp1
        k_silu<<<gsz(NAH), 256, 0, stream>>>(tmpA, z_u1[l], NAH);
        gemm(m_st[l], tmpA, lay[l][6].w, lay[l][6].b, NA, H3, HD, 0);         // umlp2
        k_update<<<gsz(NAH), 256, 0, stream>>>(m_st[l], Uv[l], Vv[l], s, v);
    }
    gemm(z1, s, r1.w, r1.b, NA, HD, HD, 0);                                   // r1
    k_silu<<<gsz(NAH), 256, 0, stream>>>(tmpA, z1, NAH);
    k_energy<<<gsz(NA), 256, 0, stream>>>(tmpA, r2.w, r2.b, image_idx, (float*)d_out);

    // ---------------------------- backward ----------------------------------
    k_dz1<<<gsz(NAH), 256, 0, stream>>>(z1, r2.w, tmpA);
    gemm(d_s, tmpA, r1.w, nullptr, NA, HD, HD, 1);                            // d_s = dz1 @ r1w^T
    hipMemsetAsync(d_va, 0, (size_t)NA * H3 * sizeof(float), stream);
    hipMemsetAsync(d_rbf, 0, (size_t)NE * RBFN * sizeof(float), stream);
    hipMemsetAsync(d_fc, 0, (size_t)NE * sizeof(float), stream);
    hipMemsetAsync(d_un, 0, (size_t)NE * 3 * sizeof(float), stream);

    for (int l = NL - 1; l >= 0; --l) {
        // update-block backward
        k_upd_bwd1<<<gsz(NAH), 256, 0, stream>>>(m_st[l], Uv[l], Vv[l],
                                                 d_s, d_va, d_m, d_Uvb, d_Vvb);
        gemm(tmpA, d_m, lay[l][6].w, nullptr, NA, HD, H3, 1);                 // d_hu
        k_silu_bwd<<<gsz(NAH), 256, 0, stream>>>(tmpA, z_u1[l], NAH);         // d_zu1
        gemm(catb, tmpA, lay[l][5].w, nullptr, NA, 256, HD, 1);               // d_cat
        k_upd_bwd2<<<gsz(NAH), 256, 0, stream>>>(Vv[l], catb, d_Vvb, d_s);
        gemm(d_va, d_Uvb, lay[l][3].w, nullptr, NA * 3, HD, HD, 1 | 2);       // d_v += dUv @ Uw^T
        gemm(d_va, d_Vvb, lay[l][4].w, nullptr, NA * 3, HD, HD, 1 | 2);       // d_v += dVv @ Vw^T
        // message-pass backward (fwlin recomputed in-register via WMMA)
        hipMemcpyAsync(d_vb, d_va, (size_t)NA * H3 * sizeof(float),
                       hipMemcpyDeviceToDevice, stream);
        hipMemsetAsync(d_h, 0, (size_t)NA * H3 * sizeof(float), stream);
        k_edge_bwd_fused<<<edgeBlocks, 256, 0, stream>>>(rbf, lay[l][0].w, lay[l][0].b,
                                                         fcut, h_st[l], v_in[l],
                                                         unit, edge, d_s, d_va, d_vb,
                                                         d_h, bufE2, d_fc, d_un);
        { float* tswap = d_va; d_va = d_vb; d_vb = tswap; }
        gemm(d_rbf, bufE2, lay[l][0].w, nullptr, NE, RBFN, H3, 1 | 2);        // d_rbf += dfwlin @ Fw^T
        gemm(tmpA, d_h, lay[l][2].w, nullptr, NA, HD, H3, 1);                 // d_hs
        k_silu_bwd<<<gsz(NAH), 256, 0, stream>>>(tmpA, z_s1[l], NAH);         // d_zs1
        gemm(d_s, tmpA, lay[l][1].w, nullptr, NA, HD, HD, 1 | 2);             // d_s += d_zs1 @ S1w^T
    }
    k_geom_bwd<<<gsz(NE), 256, 0, stream>>>(dlen, unit, d_rbf, d_fc, d_un,
                                            edge, (float*)d_out);
}